// EGATConv_39599598469257
// MI455X (gfx1250) — compile-verified
//
#include <hip/hip_runtime.h>
#include <hip/hip_bf16.h>
#include <math.h>

typedef __attribute__((ext_vector_type(2))) float v2f;
typedef __attribute__((ext_vector_type(8))) float v8f;

#define NEG_SLOPE 0.2f

__device__ __forceinline__ void atomicMaxFloat(float* addr, float val) {
    // works for mixed signs given init to -inf (0xFF800000)
    if (val >= 0.0f) atomicMax((int*)addr, __float_as_int(val));
    else             atomicMin((unsigned int*)addr, __float_as_uint(val));
}

__device__ __forceinline__ float leaky(float x) {
    return x > 0.0f ? x : NEG_SLOPE * x;
}

// ---------------- init: out=bias, m=-inf, denom/cnt/attr_sum=0 ----------------
__global__ void k_init(float* __restrict__ out, const float* __restrict__ bias,
                       float* __restrict__ m, float* __restrict__ denom,
                       float* __restrict__ cnt, float* __restrict__ attr_sum, int N) {
    long long i = (long long)blockIdx.x * blockDim.x + threadIdx.x;
    if (i < (long long)N * 128) out[i] = bias[i & 127];
    if (i < (long long)N * 16)  attr_sum[i] = 0.0f;
    if (i < (long long)N * 4) { m[i] = __int_as_float(0xFF800000); denom[i] = 0.0f; }
    if (i < (long long)N)       cnt[i] = 0.0f;
}

// ---------------- fold W_edge with att_edge: w_att[k][h], 16x4 ----------------
__global__ void k_watt(const float* __restrict__ W_edge, const float* __restrict__ att_edge,
                       float* __restrict__ w_att) {
    int t = threadIdx.x;            // 0..63
    int k = t >> 2, h = t & 3;
    float s = 0.0f;
    #pragma unroll
    for (int c = 0; c < 32; ++c)
        s += W_edge[k * 128 + h * 32 + c] * att_edge[h * 32 + c];
    w_att[k * 4 + h] = s;
}

// ---------------- in-degree + edge_attr segment sum (for self-loop fill) -----
__global__ void k_deg(const int* __restrict__ eidx, const float* __restrict__ edge_attr,
                      float* __restrict__ cnt, float* __restrict__ attr_sum, int E) {
    long long t = (long long)blockIdx.x * blockDim.x + threadIdx.x;
    if (t >= (long long)E * 16) return;
    int e = (int)(t >> 4), j = (int)(t & 15);
    int dst = eidx[E + e];
    atomicAdd(&attr_sum[(long long)dst * 16 + j], edge_attr[t]);
    if (j == 0) atomicAdd(&cnt[dst], 1.0f);
}

// ---------------- x_l = x @ W via fp32 WMMA; fused a_src/a_dst ---------------
__global__ __launch_bounds__(128)
void k_gemm(const float* __restrict__ x, const float* __restrict__ W,
            const float* __restrict__ att_src, const float* __restrict__ att_dst,
            float* __restrict__ xl, float* __restrict__ asrc, float* __restrict__ adst, int N) {
    __shared__ float sx[16 * 132];               // 16x128 A-tile, padded stride
    const int base = blockIdx.x * 16;
    const int tid  = threadIdx.x;

    for (int i = tid; i < 16 * 128; i += 128) {
        int r = i >> 7, c = i & 127;
        int row = base + r; if (row >= N) row = N - 1;
        sx[r * 132 + c] = x[(long long)row * 128 + c];
    }
    __syncthreads();

    const int w    = tid >> 5;                   // wave id == head id (4 waves)
    const int lane = tid & 31;
    const int nlow = lane & 15;
    const int hi   = lane >> 4;
    const int kk   = hi * 2;                     // K sub-pair selected by half-wave
    const int col0 = w * 32 + nlow;              // first N-tile column

    v8f c0 = {}; v8f c1 = {};

    #pragma unroll 4
    for (int k0 = 0; k0 < 128; k0 += 4) {
        v2f a;  a.x  = sx[nlow * 132 + k0 + kk];        a.y  = sx[nlow * 132 + k0 + kk + 1];
        v2f b0; b0.x = W[(k0 + kk) * 128 + col0];       b0.y = W[(k0 + kk + 1) * 128 + col0];
        v2f b1; b1.x = W[(k0 + kk) * 128 + col0 + 16];  b1.y = W[(k0 + kk + 1) * 128 + col0 + 16];
        c0 = __builtin_amdgcn_wmma_f32_16x16x4_f32(false, a, false, b0, (short)0, c0, false, false);
        c1 = __builtin_amdgcn_wmma_f32_16x16x4_f32(false, a, false, b1, (short)0, c1, false, false);
    }

    // attention vectors for this head's 32 columns
    const float as0 = att_src[w * 32 + nlow], as1 = att_src[w * 32 + 16 + nlow];
    const float ad0 = att_dst[w * 32 + nlow], ad1 = att_dst[w * 32 + 16 + nlow];

    #pragma unroll
    for (int v = 0; v < 8; ++v) {
        int row = base + v + 8 * hi;             // C/D layout: M = v + 8*(lane>=16)
        float e0 = c0[v], e1 = c1[v];
        if (row < N) {
            xl[(long long)row * 128 + col0]      = e0;
            xl[(long long)row * 128 + col0 + 16] = e1;
        }
        float ps = e0 * as0 + e1 * as1;          // per-head dot partials over N cols
        float pd = e0 * ad0 + e1 * ad1;
        #pragma unroll
        for (int msk = 1; msk < 16; msk <<= 1) { // reduce across the 16 column lanes
            ps += __shfl_xor(ps, msk, 32);
            pd += __shfl_xor(pd, msk, 32);
        }
        if (nlow == 0 && row < N) {
            asrc[row * 4 + w] = ps;
            adst[row * 4 + w] = pd;
        }
    }
}

// ---------------- per-edge logits + segment max ------------------------------
__global__ void k_edge_logits(const int* __restrict__ eidx, const float* __restrict__ edge_attr,
                              const float* __restrict__ w_att,
                              const float* __restrict__ asrc, const float* __restrict__ adst,
                              float* __restrict__ logits, float* __restrict__ m, int E) {
    __shared__ float swa[64];
    if (threadIdx.x < 64) swa[threadIdx.x] = w_att[threadIdx.x];
    __syncthreads();
    int e = blockIdx.x * blockDim.x + threadIdx.x;
    if (e >= E) return;
    int src = eidx[e], dst = eidx[E + e];
    const float* ea = edge_attr + (long long)e * 16;
    float a0 = 0.f, a1 = 0.f, a2 = 0.f, a3 = 0.f;
    #pragma unroll
    for (int k = 0; k < 16; ++k) {
        float v = ea[k];
        a0 += v * swa[k * 4 + 0]; a1 += v * swa[k * 4 + 1];
        a2 += v * swa[k * 4 + 2]; a3 += v * swa[k * 4 + 3];
    }
    float ae[4] = {a0, a1, a2, a3};
    #pragma unroll
    for (int h = 0; h < 4; ++h) {
        float l = leaky(asrc[src * 4 + h] + adst[dst * 4 + h] + ae[h]);
        logits[(long long)e * 4 + h] = l;
        atomicMaxFloat(&m[dst * 4 + h], l);
    }
}

// ---------------- self-loop logits (fill_value='mean') + segment max ---------
__global__ void k_loop_logits(const float* __restrict__ cnt, const float* __restrict__ attr_sum,
                              const float* __restrict__ w_att,
                              const float* __restrict__ asrc, const float* __restrict__ adst,
                              float* __restrict__ logits, float* __restrict__ m, int E, int N) {
    __shared__ float swa[64];
    if (threadIdx.x < 64) swa[threadIdx.x] = w_att[threadIdx.x];
    __syncthreads();
    int n = blockIdx.x * blockDim.x + threadIdx.x;
    if (n >= N) return;
    float inv = 1.0f / fmaxf(cnt[n], 1.0f);
    float a0 = 0.f, a1 = 0.f, a2 = 0.f, a3 = 0.f;
    #pragma unroll
    for (int k = 0; k < 16; ++k) {
        float v = attr_sum[(long long)n * 16 + k] * inv;
        a0 += v * swa[k * 4 + 0]; a1 += v * swa[k * 4 + 1];
        a2 += v * swa[k * 4 + 2]; a3 += v * swa[k * 4 + 3];
    }
    float ae[4] = {a0, a1, a2, a3};
    #pragma unroll
    for (int h = 0; h < 4; ++h) {
        float l = leaky(asrc[n * 4 + h] + adst[n * 4 + h] + ae[h]);
        logits[(long long)(E + n) * 4 + h] = l;
        atomicMaxFloat(&m[n * 4 + h], l);
    }
}

// ---------------- softmax denominator ----------------------------------------
__global__ void k_denom(const int* __restrict__ eidx, const float* __restrict__ logits,
                        const float* __restrict__ m, float* __restrict__ denom, int E, int N) {
    int e = blockIdx.x * blockDim.x + threadIdx.x;
    if (e >= E + N) return;
    int dst = (e < E) ? eidx[E + e] : (e - E);
    #pragma unroll
    for (int h = 0; h < 4; ++h) {
        float p = expf(logits[(long long)e * 4 + h] - m[dst * 4 + h]);
        atomicAdd(&denom[dst * 4 + h], p);
    }
}

// ---------------- weighted scatter: out[dst] += alpha * x_l[src] -------------
__global__ void k_scatter(const int* __restrict__ eidx, const float* __restrict__ logits,
                          const float* __restrict__ m, const float* __restrict__ denom,
                          const float* __restrict__ xl, float* __restrict__ out, int E, int N) {
    long long t = (long long)blockIdx.x * blockDim.x + threadIdx.x;
    if (t >= (long long)(E + N) * 32) return;
    int e    = (int)(t >> 5);
    int lane = (int)(t & 31);
    int src, dst;
    if (e < E) { src = eidx[e]; dst = eidx[E + e]; } else { src = dst = e - E; }
    int h = lane >> 3;                                   // 4 floats per lane -> head = lane/8
    float alpha = expf(logits[(long long)e * 4 + h] - m[dst * 4 + h]) / denom[dst * 4 + h];
    const float4 v = *reinterpret_cast<const float4*>(xl + (long long)src * 128 + lane * 4);
    float* o = out + (long long)dst * 128 + lane * 4;
    atomicAdd(o + 0, alpha * v.x);
    atomicAdd(o + 1, alpha * v.y);
    atomicAdd(o + 2, alpha * v.z);
    atomicAdd(o + 3, alpha * v.w);
}

extern "C" void kernel_launch(void* const* d_in, const int* in_sizes, int n_in,
                              void* d_out, int out_size, void* d_ws, size_t ws_size,
                              hipStream_t stream) {
    const float* x        = (const float*)d_in[0];
    const int*   eidx     = (const int*)d_in[1];
    const float* edge_attr= (const float*)d_in[2];
    const float* W        = (const float*)d_in[3];
    const float* W_edge   = (const float*)d_in[4];
    const float* att_src  = (const float*)d_in[5];
    const float* att_dst  = (const float*)d_in[6];
    const float* att_edge = (const float*)d_in[7];
    const float* bias     = (const float*)d_in[8];
    float* out = (float*)d_out;

    const int N = in_sizes[0] / 128;
    const int E = in_sizes[1] / 2;

    float* ws = (float*)d_ws;
    float* xl       = ws; ws += (size_t)N * 128;
    float* asrc     = ws; ws += (size_t)N * 4;
    float* adst     = ws; ws += (size_t)N * 4;
    float* cnt      = ws; ws += (size_t)N;
    float* attr_sum = ws; ws += (size_t)N * 16;
    float* mbuf     = ws; ws += (size_t)N * 4;
    float* denom    = ws; ws += (size_t)N * 4;
    float* w_att    = ws; ws += 64;
    float* logits   = ws; ws += (size_t)(E + N) * 4;

    const int T = 256;
    {   // init
        long long n = (long long)N * 128;
        k_init<<<(unsigned)((n + T - 1) / T), T, 0, stream>>>(out, bias, mbuf, denom, cnt, attr_sum, N);
    }
    k_watt<<<1, 64, 0, stream>>>(W_edge, att_edge, w_att);
    {   // degree + attr segment sum
        long long n = (long long)E * 16;
        k_deg<<<(unsigned)((n + T - 1) / T), T, 0, stream>>>(eidx, edge_attr, cnt, attr_sum, E);
    }
    k_gemm<<<(unsigned)((N + 15) / 16), 128, 0, stream>>>(x, W, att_src, att_dst, xl, asrc, adst, N);
    k_edge_logits<<<(unsigned)((E + T - 1) / T), T, 0, stream>>>(eidx, edge_attr, w_att, asrc, adst,
                                                                 logits, mbuf, E);
    k_loop_logits<<<(unsigned)((N + T - 1) / T), T, 0, stream>>>(cnt, attr_sum, w_att, asrc, adst,
                                                                 logits, mbuf, E, N);
    k_denom<<<(unsigned)((E + N + T - 1) / T), T, 0, stream>>>(eidx, logits, mbuf, denom, E, N);
    {   // scatter
        long long n = (long long)(E + N) * 32;
        k_scatter<<<(unsigned)((n + T - 1) / T), T, 0, stream>>>(eidx, logits, mbuf, denom, xl, out, E, N);
    }
}